// GraphConvolution_84885733638259
// MI455X (gfx1250) — compile-verified
//
#include <hip/hip_runtime.h>

typedef __attribute__((ext_vector_type(2))) float v2f;
typedef __attribute__((ext_vector_type(8))) float v8f;

// ---------------------------------------------------------------------------
// Kernel 1: zero the workspace (AX accumulator + degree accumulator)
// ---------------------------------------------------------------------------
__global__ void gcn_zero_kernel(float* __restrict__ p, size_t n) {
    size_t i = (size_t)blockIdx.x * blockDim.x + threadIdx.x;
    size_t stride = (size_t)gridDim.x * blockDim.x;
    for (; i < n; i += stride) p[i] = 0.0f;
}

// ---------------------------------------------------------------------------
// Kernel 2: edge scatter.  One wave32 per directed edge.
//   AX[dst] += w * X[src]   (128 floats, 4 per lane via float4 load)
//   deg[dst] += w           (lane 0)
// AX is 25.6 MB -> atomics resolve in the 192 MB L2.
// ---------------------------------------------------------------------------
__global__ void gcn_scatter_kernel(const float* __restrict__ X,
                                   const int* __restrict__ edges,
                                   const float* __restrict__ wts,
                                   float* __restrict__ AX,
                                   float* __restrict__ deg,
                                   int nDirected /* = 2*E */) {
    int gid  = blockIdx.x * blockDim.x + threadIdx.x;
    int wave = gid >> 5;
    int lane = gid & 31;
    if (wave >= nDirected) return;

    int e   = wave >> 1;
    int dir = wave & 1;
    int src = edges[2 * e + dir];
    int dst = edges[2 * e + (dir ^ 1)];
    float w = wts[e];

    const float4* xrow = (const float4*)(X + (size_t)src * 128);
    float4 v = xrow[lane];                      // 4 contiguous floats per lane

    float* o = AX + (size_t)dst * 128 + lane * 4;
    atomicAdd(o + 0, w * v.x);
    atomicAdd(o + 1, w * v.y);
    atomicAdd(o + 2, w * v.z);
    atomicAdd(o + 3, w * v.w);

    if (lane == 0) atomicAdd(deg + dst, w);
}

// ---------------------------------------------------------------------------
// Kernel 3: out = (AX * inv_deg) @ W via V_WMMA_F32_16X16X4_F32.
// One wave per 16-row strip; 8 f32 accumulators cover all 128 output cols.
// A 16x4 f32 layout (ISA 7.12.2): lanes 0-15 -> K=0,1 ; lanes 16-31 -> K=2,3
// B  4x16 layout: VGPR0 = K-rows {0,2}, VGPR1 = K-rows {1,3} (lane halves)
// C/D 16x16:      VGPR r = row r (lanes 0-15) / row r+8 (lanes 16-31)
// ---------------------------------------------------------------------------
__global__ void gcn_gemm_kernel(const float* __restrict__ AX,
                                const float* __restrict__ deg,
                                const float* __restrict__ W,   // [128,128] row-major
                                float* __restrict__ out,        // [N,128]
                                int mtiles /* = N/16 */) {
    int lane  = threadIdx.x & 31;
    int mtile = blockIdx.x * (blockDim.x >> 5) + (threadIdx.x >> 5);
    if (mtile >= mtiles) return;   // wave-uniform: EXEC stays all-ones below

    int half   = lane >> 4;        // 0: lanes 0-15, 1: lanes 16-31
    int mlocal = lane & 15;
    int row    = mtile * 16 + mlocal;

    float d   = deg[row];
    float inv = (d > 0.0f) ? (1.0f / d) : 0.0f;
    const float* yrow = AX + (size_t)row * 128;

    v8f acc[8];
    #pragma unroll
    for (int n = 0; n < 8; ++n) acc[n] = (v8f){0,0,0,0,0,0,0,0};

    #pragma unroll 4
    for (int k = 0; k < 128; k += 4) {
        int kb = k + half * 2;     // this lane's pair of K values
        v2f a;
        a.x = yrow[kb]     * inv;
        a.y = yrow[kb + 1] * inv;
        #pragma unroll
        for (int n = 0; n < 8; ++n) {
            int col = n * 16 + mlocal;
            v2f b;
            b.x = W[(size_t)kb * 128 + col];
            b.y = W[(size_t)(kb + 1) * 128 + col];
            acc[n] = __builtin_amdgcn_wmma_f32_16x16x4_f32(
                false, a, false, b, (short)0, acc[n], false, false);
        }
    }

    // Write back: VGPR r of each acc holds row (r + 8*half) of the tile.
    #pragma unroll
    for (int n = 0; n < 8; ++n) {
        #pragma unroll
        for (int r = 0; r < 8; ++r) {
            int orow = mtile * 16 + r + 8 * half;
            out[(size_t)orow * 128 + n * 16 + mlocal] = acc[n][r];
        }
    }
}

// ---------------------------------------------------------------------------
extern "C" void kernel_launch(void* const* d_in, const int* in_sizes, int n_in,
                              void* d_out, int out_size, void* d_ws, size_t ws_size,
                              hipStream_t stream) {
    const float* X     = (const float*)d_in[0];   // [N,128]
    const int*   edges = (const int*)d_in[1];     // [E,2]
    const float* wts   = (const float*)d_in[2];   // [E]
    const float* W     = (const float*)d_in[3];   // [128,128]
    float*       out   = (float*)d_out;

    const int N = in_sizes[0] / 128;
    const int E = in_sizes[1] / 2;

    float* AX  = (float*)d_ws;                    // N*128 floats
    float* deg = AX + (size_t)N * 128;            // N floats

    // 1) zero AX + deg
    size_t zn = (size_t)N * 128 + (size_t)N;
    gcn_zero_kernel<<<2048, 256, 0, stream>>>(AX, zn);

    // 2) scatter: one wave per directed edge
    int nDirected = 2 * E;
    long long threads = (long long)nDirected * 32;
    int blocks = (int)((threads + 255) / 256);
    gcn_scatter_kernel<<<blocks, 256, 0, stream>>>(X, edges, wts, AX, deg, nDirected);

    // 3) WMMA GEMM: one wave per 16-row strip, 8 waves per block
    int mtiles = N / 16;                          // N=50000 -> 3125, exact
    int wavesPerBlock = 8;
    int gblocks = (mtiles + wavesPerBlock - 1) / wavesPerBlock;
    gcn_gemm_kernel<<<gblocks, wavesPerBlock * 32, 0, stream>>>(AX, deg, W, out, mtiles);
}